// CGGRLoss_25383256720133
// MI455X (gfx1250) — compile-verified
//
#include <hip/hip_runtime.h>
#include <hip/hip_bf16.h>
#include <math.h>
#include <stdint.h>

// Problem constants (from reference): B=2, S=2048, V=50257
#define VDIM   50257
#define NROWS  4096            // B*S
#define NT1    512             // threads per row block (16 wave32)
#define CHUNK  (NT1 * 4)       // 2048 floats staged per chunk (16B per lane)
#define NBUF   4               // ring depth: 3 chunks in flight per wave
#define NFULL  (VDIM / CHUNK)  // 24 full chunks = 49152 floats
#define TAILB  (NFULL * CHUNK)
#define NT2    1024

// ---- CDNA5 async global->LDS copy (ASYNCcnt-tracked), per-lane 16B ----
#define ASYNC_LOAD_B128(ldsoff, gaddr)                                        \
  asm volatile("global_load_async_to_lds_b128 %0, %1, off"                    \
               :: "v"(ldsoff), "v"(gaddr) : "memory")
#define WAIT_ASYNCCNT(n)                                                      \
  asm volatile("s_wait_asynccnt %0" :: "n"(n) : "memory")
#define WAIT_DSCNT0()                                                         \
  asm volatile("s_wait_dscnt 0x0" ::: "memory")

// Branchless online-softmax merge of one scalar into (m, s, w, t2)
__device__ __forceinline__ void merge1(float x, float& m, float& s, float& w, float& t2) {
  float mo = m;
  m  = fmaxf(mo, x);
  t2 = fmaxf(t2, fminf(mo, x));
  float r = __expf(mo - m);     // 1.0 when max unchanged
  float e = __expf(x  - m);     // 1.0 when x is the new max
  s = s * r + e;
  w = w * r + x * e;
}

// Merge a 4-group: one rescale exp + four element exps, branchless top-2.
__device__ __forceinline__ void merge4(float4 v, float& m, float& s, float& w, float& t2) {
  const float x0 = v.x, x1 = v.y, x2 = v.z, x3 = v.w;
  float ma = fmaxf(x0, x1), na = fminf(x0, x1);
  float mb = fmaxf(x2, x3), nb = fminf(x2, x3);
  float cmax = fmaxf(ma, mb);
  float csec = fmaxf(fminf(ma, mb), (ma >= mb) ? na : nb);
  float mo = m;
  m  = fmaxf(mo, cmax);
  t2 = fmaxf(fminf(mo, cmax), (mo >= cmax) ? t2 : csec);
  float r  = __expf(mo - m);
  float e0 = __expf(x0 - m), e1 = __expf(x1 - m);
  float e2 = __expf(x2 - m), e3 = __expf(x3 - m);
  s = s * r + ((e0 + e1) + (e2 + e3));
  w = w * r + ((x0 * e0 + x1 * e1) + (x2 * e2 + x3 * e3));
}

// Kernel 1: one block per row; single-pass online softmax + top2 + target pick.
__global__ __launch_bounds__(NT1)
void cggr_rowstats(const float* __restrict__ logits,
                   const int*   __restrict__ targets,
                   float* __restrict__ d_loss,
                   float* __restrict__ d_conf,
                   float* __restrict__ d_diff)
{
  __shared__ __align__(16) float stage[NBUF][CHUNK];  // 32 KB async staging ring
  __shared__ float red_m[NT1], red_s[NT1], red_w[NT1], red_t2[NT1];

  const int row = blockIdx.x;
  const int tid = threadIdx.x;
  const float* rowp = logits + (size_t)row * (size_t)VDIM;

  // One lane fetches the target logit early (latency hidden under the stream).
  float x_t = 0.0f;
  if (tid == 0) x_t = rowp[targets[row]];

  const int tid4 = tid * 4;
  const uint32_t ldsbase = (uint32_t)(uintptr_t)(&stage[0][tid4]);

  // Prime the pipeline: chunks 0..2 -> buffers 0..2 (ASYNCcnt = 3 per wave)
  #pragma unroll
  for (int p = 0; p < NBUF - 1; ++p) {
    ASYNC_LOAD_B128(ldsbase + (uint32_t)p * (CHUNK * 4),
                    (uint64_t)(uintptr_t)(rowp + (size_t)p * CHUNK + tid4));
  }

  float m = -INFINITY, t2 = -INFINITY, s = 0.0f, w = 0.0f;

  // Steady state: keep 3 chunks in flight; wait for the oldest, consume it.
  for (int c = 0; c + (NBUF - 1) < NFULL; ++c) {
    // Buffer being refilled was consumed last iteration; ensure its ds_read
    // fully retired before the async engine may overwrite it (WAR hazard).
    WAIT_DSCNT0();
    const int cn = c + (NBUF - 1);
    ASYNC_LOAD_B128(ldsbase + (uint32_t)(cn & (NBUF - 1)) * (CHUNK * 4),
                    (uint64_t)(uintptr_t)(rowp + (size_t)cn * CHUNK + tid4));
    WAIT_ASYNCCNT(NBUF - 1);    // oldest chunk (c) complete: async retires in order
    merge4(*(const float4*)&stage[c & (NBUF - 1)][tid4], m, s, w, t2);
  }
  // Drain: chunks NFULL-3, NFULL-2, NFULL-1
  WAIT_ASYNCCNT(2);
  merge4(*(const float4*)&stage[(NFULL - 3) & (NBUF - 1)][tid4], m, s, w, t2);
  WAIT_ASYNCCNT(1);
  merge4(*(const float4*)&stage[(NFULL - 2) & (NBUF - 1)][tid4], m, s, w, t2);
  WAIT_ASYNCCNT(0);
  merge4(*(const float4*)&stage[(NFULL - 1) & (NBUF - 1)][tid4], m, s, w, t2);

  // Tail: 50257 - 49152 = 1105 elements, guarded direct loads.
  for (int i = TAILB + tid; i < VDIM; i += NT1) {
    merge1(rowp[i], m, s, w, t2);
  }

  // Block reduction with the exact online-softmax merge rule.
  red_m[tid] = m; red_s[tid] = s; red_w[tid] = w; red_t2[tid] = t2;
  __syncthreads();
  for (int off = NT1 / 2; off > 0; off >>= 1) {
    if (tid < off) {
      float ma2 = red_m[tid],        sa = red_s[tid];
      float wa  = red_w[tid],        ta = red_t2[tid];
      float mb2 = red_m[tid + off],  sb = red_s[tid + off];
      float wb  = red_w[tid + off],  tb = red_t2[tid + off];
      float mh = fmaxf(ma2, mb2);
      float ea = __expf(ma2 - mh), eb = __expf(mb2 - mh);
      red_m[tid]  = mh;
      red_s[tid]  = sa * ea + sb * eb;
      red_w[tid]  = wa * ea + wb * eb;
      red_t2[tid] = fmaxf(fminf(ma2, mb2), (ma2 >= mb2) ? ta : tb);
    }
    __syncthreads();
  }

  if (tid == 0) {
    const float mf = red_m[0], sf = red_s[0], wf = red_w[0], t2f = red_t2[0];
    const float logZ    = mf + logf(sf);
    const float entropy = logZ - wf / sf;          // logZ - E_p[x]
    const float conf    = __expf(mf  - logZ);      // top-1 probability
    const float p2      = __expf(t2f - logZ);      // top-2 probability
    const float margin  = conf - p2;
    const float loss    = logZ - x_t;
    const float inv_logV = 1.0f / logf((float)VDIM);
    const float diff = (entropy * inv_logV + (1.0f - margin) + loss * inv_logV)
                       * (1.0f / 3.0f);
    d_loss[row] = loss;
    d_conf[row] = conf;
    d_diff[row] = diff;
  }
}

// Kernel 2: single block. mean_conf -> k; exact k-th largest via bitonic sort
// of the 4096 difficulties in LDS; masked mean of per-token losses.
__global__ __launch_bounds__(NT2)
void cggr_finalize(const float* __restrict__ d_diff,
                   const float* __restrict__ d_loss,
                   const float* __restrict__ d_conf,
                   const int*   __restrict__ step_count,
                   float* __restrict__ out)
{
  __shared__ float sortbuf[NROWS];   // 16 KB
  __shared__ float redA[NT2];        // 4 KB
  __shared__ float redB[NT2];        // 4 KB
  __shared__ int   s_k;

  const int tid = threadIdx.x;

  // Load difficulties into sort buffer, accumulate confidence sum.
  float csum = 0.0f;
  for (int i = tid; i < NROWS; i += NT2) {
    sortbuf[i] = d_diff[i];
    csum += d_conf[i];
  }
  redA[tid] = csum;
  __syncthreads();
  for (int off = NT2 / 2; off > 0; off >>= 1) {
    if (tid < off) redA[tid] += redA[tid + off];
    __syncthreads();
  }

  if (tid == 0) {
    const float mean_conf = redA[0] / (float)NROWS;
    const float progress  = fminf(1.0f, (float)step_count[0] / 1000.0f);
    const float base      = 1.0f - progress * (1.0f - 0.25f);
    float ratio = base * (1.0f + 0.5f * (0.5f - mean_conf));
    ratio = fminf(fmaxf(ratio, 0.05f), 1.0f);
    int k = (int)rintf(ratio * (float)NROWS);      // round-half-even like jnp.round
    k = min(max(k, 1), NROWS);
    s_k = k;
  }
  __syncthreads();

  // Bitonic sort ascending (exact selection of the k-th largest).
  for (int kk = 2; kk <= NROWS; kk <<= 1) {
    for (int j = kk >> 1; j > 0; j >>= 1) {
      for (int i = tid; i < NROWS; i += NT2) {
        const int ij = i ^ j;
        if (ij > i) {
          const float a = sortbuf[i], b = sortbuf[ij];
          const bool asc = ((i & kk) == 0);
          if ((a > b) == asc) { sortbuf[i] = b; sortbuf[ij] = a; }
        }
      }
      __syncthreads();
    }
  }

  const float thresh = sortbuf[NROWS - s_k];       // k-th largest

  float lsum = 0.0f, cnt = 0.0f;
  for (int i = tid; i < NROWS; i += NT2) {
    if (d_diff[i] >= thresh) { lsum += d_loss[i]; cnt += 1.0f; }
  }
  redA[tid] = lsum;
  redB[tid] = cnt;
  __syncthreads();
  for (int off = NT2 / 2; off > 0; off >>= 1) {
    if (tid < off) { redA[tid] += redA[tid + off]; redB[tid] += redB[tid + off]; }
    __syncthreads();
  }

  if (tid == 0) out[0] = redA[0] / fmaxf(redB[0], 1.0f);
}

extern "C" void kernel_launch(void* const* d_in, const int* in_sizes, int n_in,
                              void* d_out, int out_size, void* d_ws, size_t ws_size,
                              hipStream_t stream) {
  const float* logits  = (const float*)d_in[0];
  const int*   targets = (const int*)d_in[1];   // per-doc contract: integer -> const int*
  const int*   step    = (const int*)d_in[2];   // low dword holds the value either way
  float* out = (float*)d_out;

  float* ws     = (float*)d_ws;
  float* d_loss = ws;                 // [4096]
  float* d_conf = ws + NROWS;         // [4096]
  float* d_diff = ws + 2 * NROWS;     // [4096]

  hipLaunchKernelGGL(cggr_rowstats, dim3(NROWS), dim3(NT1), 0, stream,
                     logits, targets, d_loss, d_conf, d_diff);
  hipLaunchKernelGGL(cggr_finalize, dim3(1), dim3(NT2), 0, stream,
                     d_diff, d_loss, d_conf, step, out);
}